// MSErrLossAll2_32323923870520
// MI455X (gfx1250) — compile-verified
//
#include <hip/hip_runtime.h>
#include <hip/hip_bf16.h>

typedef __attribute__((ext_vector_type(2))) float v2f;
typedef __attribute__((ext_vector_type(8))) float v8f;

#define BLOCK   256
#define NBLOCKS 2048

// log(I0(x)) for x >= 0, Abramowitz & Stegun 9.8.1 / 9.8.2 (~1e-7 rel).
__device__ __forceinline__ float log_i0(float x) {
  float t  = x * (1.0f / 3.75f);
  float t2 = t * t;
  float pSmall = 1.0f + t2 * (3.5156229f + t2 * (3.0899424f + t2 * (1.2067492f +
                 t2 * (0.2659732f + t2 * (0.0360768f + t2 * 0.0045813f)))));
  float small = __logf(pSmall);
  float ti = 1.0f / t; // == 3.75/x
  float pBig = 0.39894228f + ti * (0.01328592f + ti * (0.00225319f + ti * (-0.00157565f +
               ti * (0.00916281f + ti * (-0.02057706f + ti * (0.02635537f +
               ti * (-0.01647633f + ti * 0.00392377f)))))));
  float big = x - 0.5f * __logf(x) + __logf(pBig);
  return (x < 3.75f) ? small : big;
}

__device__ __forceinline__ float row_loss(const float* __restrict__ ir,
                                          const float* __restrict__ tr) {
  float x0 = ir[0], x1 = ir[1], x2 = ir[2], x3 = ir[3], x4 = ir[4], x5 = ir[5];
  float y0 = tr[0], y1 = tr[1], y2 = tr[2], y3 = tr[3], y4 = tr[4];
  float rinv = rsqrtf(x0 * x0 + x1 * x1);
  float n0 = x0 * rinv, n1 = x1 * rinv;
  float kappa = __expf(-x2);
  float align = (n0 * n0 - n1 * n1) * (y0 * y0 - y1 * y1) + 4.0f * n0 * n1 * y0 * y1;
  float d0 = n0 - y0, d1 = n1 - y1;
  float e3 = x3 - y2, e4 = x4 - y3, e5 = x5 - y4;
  return 0.5f * (d0 * d0 + d1 * d1)     // ALPHA = 1
       - kappa * align
       + (e3 * e3 + e4 * e4)            // LAMBDA_ABS = 1
       + 0.4f * (e5 * e5)               // 0.4 * LAMBDA_E
       + log_i0(kappa);
}

// Full-wave (32-lane) sum via V_WMMA_F32_16X16X4_F32 with B = ones.
// A layout (16x4 f32): lanes 0-15 hold K=0,1 in VGPR0,1; lanes 16-31 hold K=2,3.
// With a = {acc, 0}: D[m][n] = acc[m] + acc[m+16] for every column n.
// D layout: lane L<16 VGPR v -> (M=v, N=L); lane L>=16 -> (M=v+8, N=L-16).
// So sum of one lane's 8 D regs = half-sum; xor-16 shuffle adds the other half.
// Requires EXEC all ones (call with no divergence).
__device__ __forceinline__ float wave_reduce_wmma(float acc) {
  v2f a; a[0] = acc;  a[1] = 0.0f;
  v2f b; b[0] = 1.0f; b[1] = 1.0f;
  v8f c = {};
  c = __builtin_amdgcn_wmma_f32_16x16x4_f32(false, a, false, b,
                                            (short)0, c, false, false);
  float s = c[0] + c[1] + c[2] + c[3] + c[4] + c[5] + c[6] + c[7];
  s += __shfl_xor(s, 16, 32);
  return s; // full wave sum in every lane
}

__global__ void loss_partial_kernel(const float* __restrict__ in,
                                    const float* __restrict__ tg,
                                    float* __restrict__ partial,
                                    int nTiles, int nRows) {
  __shared__ float s_in[BLOCK * 6];   // 6 KB
  __shared__ float s_tg[BLOCK * 5];   // 5 KB
  __shared__ float s_wave[BLOCK / 32];

  const int t = threadIdx.x;
  float acc = 0.0f;

  for (int tile = blockIdx.x; tile < nTiles; tile += gridDim.x) {
    const float4* inT = (const float4*)(in + (size_t)tile * (BLOCK * 6));
    const float4* tgT = (const float4*)(tg + (size_t)tile * (BLOCK * 5));

    // Prefetch next tile's cachelines (global_prefetch_b8).
    int nt = tile + gridDim.x;
    if (nt < nTiles) {
      const char* pin = (const char*)(in + (size_t)nt * (BLOCK * 6)); // 6144 B = 48 lines
      const char* ptg = (const char*)(tg + (size_t)nt * (BLOCK * 5)); // 5120 B = 40 lines
      if (t < 48)      __builtin_prefetch(pin + t * 128, 0, 0);
      else if (t < 88) __builtin_prefetch(ptg + (t - 48) * 128, 0, 0);
    }

    // Coalesced b128 global -> LDS staging.
    ((float4*)s_in)[t] = inT[t];                        // 256 of 384
    if (t < 128) ((float4*)s_in)[BLOCK + t] = inT[BLOCK + t];
    ((float4*)s_tg)[t] = tgT[t];                        // 256 of 320
    if (t < 64)  ((float4*)s_tg)[BLOCK + t] = tgT[BLOCK + t];
    __syncthreads();

    // Conflict-free strided reads (gcd(6,64)=2 -> 6L mod 64 distinct over 32 lanes;
    // gcd(5,64)=1 likewise).
    acc += row_loss(s_in + t * 6, s_tg + t * 5);
    __syncthreads();
  }

  // Tail rows (N % 256 != 0); N = 2^23 in the reference so normally empty.
  int rem = nRows - nTiles * BLOCK;
  if (blockIdx.x == 0 && t < rem) {
    int r = nTiles * BLOCK + t;
    acc += row_loss(in + (size_t)r * 6, tg + (size_t)r * 5);
  }

  // Wave reduce on the matrix unit, then block reduce (deterministic order).
  float s = wave_reduce_wmma(acc);
  if ((t & 31) == 0) s_wave[t >> 5] = s;
  __syncthreads();
  if (t == 0) {
    float tot = 0.0f;
#pragma unroll
    for (int i = 0; i < BLOCK / 32; ++i) tot += s_wave[i];
    partial[blockIdx.x] = tot;
  }
}

__global__ void final_reduce_kernel(const float* __restrict__ partial,
                                    float* __restrict__ out,
                                    int n, float invN) {
  __shared__ float s_wave[BLOCK / 32];
  float acc = 0.0f;
  for (int i = threadIdx.x; i < n; i += BLOCK) acc += partial[i];
  float s = wave_reduce_wmma(acc);
  if ((threadIdx.x & 31) == 0) s_wave[threadIdx.x >> 5] = s;
  __syncthreads();
  if (threadIdx.x == 0) {
    float tot = 0.0f;
#pragma unroll
    for (int i = 0; i < BLOCK / 32; ++i) tot += s_wave[i];
    out[0] = tot * invN;
  }
}

extern "C" void kernel_launch(void* const* d_in, const int* in_sizes, int n_in,
                              void* d_out, int out_size, void* d_ws, size_t ws_size,
                              hipStream_t stream) {
  const float* in = (const float*)d_in[0];   // (N, 6) float32
  const float* tg = (const float*)d_in[1];   // (N, 5) float32
  float* out = (float*)d_out;                // scalar float32 mean
  float* ws  = (float*)d_ws;                 // block partials

  long long nRows = (long long)in_sizes[0] / 6;
  int nTiles = (int)(nRows / BLOCK);
  int nb = NBLOCKS;
  if (nTiles > 0 && nb > nTiles) nb = nTiles;
  if (nb < 1) nb = 1;

  loss_partial_kernel<<<nb, BLOCK, 0, stream>>>(in, tg, ws, nTiles, (int)nRows);
  final_reduce_kernel<<<1, BLOCK, 0, stream>>>(ws, out, nb, 1.0f / (float)nRows);
}